// PatchEmbadding_39668317946555
// MI455X (gfx1250) — compile-verified
//
#include <hip/hip_runtime.h>
#include <hip/hip_bf16.h>

// out[b, ph*14+pw, c*256+i*16+j] = in[b, c, ph*16+i, pw*16+j]
// B=256, C=3, H=W=224, PATCH=16 -> out (256, 196, 768), all fp32.
//
// One block per (b, ph) strip:
//   input  : 3 contiguous chunks of 16*224 floats  (coalesced b128 loads)
//   output : 14*768 contiguous floats              (coalesced b128 stores)
// LDS stages the permutation via gfx1250 async global<->LDS b128 ops
// (ASYNCcnt path). All per-iteration address deltas are compile-time
// constants; per-thread decomposition is computed once (1 div + shifts).

#define NPH     14
#define LDSROW  228              // 224 + 4 pad words to spread LDS banks
#define LDSC    (16 * LDSROW)    // 3648 words per channel
#define LDSTOT  (3 * LDSC)       // 10944 words = 43776 bytes

#if defined(__has_builtin)
#if __has_builtin(__builtin_amdgcn_global_load_async_to_lds_b128)
#define HAVE_ASYNC_LD 1
#endif
#if __has_builtin(__builtin_amdgcn_global_store_async_from_lds_b128)
#define HAVE_ASYNC_ST 1
#endif
#if __has_builtin(__builtin_amdgcn_s_wait_asynccnt)
#define HAVE_WAIT_ASYNC 1
#endif
#endif

typedef int v4i __attribute__((ext_vector_type(4)));
typedef __attribute__((address_space(1))) v4i gv4i;   // global int4
typedef __attribute__((address_space(3))) v4i lv4i;   // LDS int4

__device__ __forceinline__ void wait_async_zero() {
#if defined(HAVE_WAIT_ASYNC)
    __builtin_amdgcn_s_wait_asynccnt(0);
#elif defined(HAVE_ASYNC_LD) || defined(HAVE_ASYNC_ST)
    asm volatile("s_wait_asynccnt 0" ::: "memory");
#endif
}

__global__ __launch_bounds__(224) void patch_embed_permute(
    const float* __restrict__ in, float* __restrict__ out) {
    __shared__ float lds[LDSTOT];

    const int tid = threadIdx.x;        // 0..223
    const int gid = blockIdx.x;         // 0..3583
    const int b   = gid / NPH;
    const int ph  = gid - b * NPH;

    // ---------------- Phase 1: global -> LDS (input-linear, coalesced) ----
    // Iteration k covers 4 input rows; c = k>>2 and row = 4*(k&3)+ib, so all
    // per-k address deltas are compile-time constants.
    {
        const int ib = tid / 56;                 // row-within-4 group (0..3)
        const int w4 = tid - ib * 56;            // float4 within 224-row
        const float* gbase = in + (size_t)b * 150528 + ph * 3584
                                + ib * 224 + w4 * 4;
        float* lbase = &lds[ib * LDSROW + w4 * 4];
#pragma unroll
        for (int k = 0; k < 12; ++k) {
            const int c  = k >> 2;
            const int kr = k & 3;
            const float* gp = gbase + c * 50176 + kr * 896;       // floats
            float*       lp = lbase + c * LDSC  + kr * (4 * LDSROW);
#if defined(HAVE_ASYNC_LD)
            __builtin_amdgcn_global_load_async_to_lds_b128(
                (gv4i*)(void*)const_cast<float*>(gp),
                (lv4i*)(void*)lp, 0, 0);
#else
            const float4 v = *(const float4*)gp;
            *(float4*)lp = v;
#endif
        }
    }

    wait_async_zero();
    __syncthreads();

    // ---------------- Phase 2: LDS -> global (output-linear, coalesced) ---
    // Each thread owns patch pw = tid>>4, column slot v = tid&15; per-k the
    // output advances by 64 floats and the LDS source by constant deltas.
    {
        const int pw = tid >> 4;                 // 0..13
        const int v  = tid & 15;                 // 0..15
        const int vi = v >> 2;                   // i sub-index (0..3)
        const int j  = (v & 3) << 2;             // 0,4,8,12
        float* outbase = out + ((size_t)b * 196 + (size_t)ph * NPH) * 768
                             + pw * 768 + v * 4;
        const float* lrbase = &lds[vi * LDSROW + pw * 16 + j];
#pragma unroll
        for (int k = 0; k < 12; ++k) {
            const int c  = k >> 2;
            const int kr = k & 3;
            const float* lp = lrbase + c * LDSC + kr * (4 * LDSROW);
            float*       gp = outbase + k * 64;
#if defined(HAVE_ASYNC_ST)
            __builtin_amdgcn_global_store_async_from_lds_b128(
                (gv4i*)(void*)gp,
                (lv4i*)(void*)const_cast<float*>(lp), 0, 0);
#else
            const float4 val = *(const float4*)lp;
            *(float4*)gp = val;
#endif
        }
    }
    // S_ENDPGM performs an implicit wait-idle, draining the async stores.
}

extern "C" void kernel_launch(void* const* d_in, const int* in_sizes, int n_in,
                              void* d_out, int out_size, void* d_ws, size_t ws_size,
                              hipStream_t stream) {
    (void)in_sizes; (void)n_in; (void)d_ws; (void)ws_size; (void)out_size;
    const float* in  = (const float*)d_in[0];
    float*       out = (float*)d_out;
    // 256 batches * 14 ph-strips = 3584 blocks, 224 threads (7 waves) each.
    patch_embed_permute<<<dim3(256 * NPH), dim3(224), 0, stream>>>(in, out);
}